// _LoRA_qkv_timm_train_61761629716785
// MI455X (gfx1250) — compile-verified
//
#include <hip/hip_runtime.h>
#include <math.h>
#include <stdint.h>

// Problem constants (from reference): B=32,S=512,D=768,R=16,T=10
#define D_MODEL 768
#define N_OUT   2304          // 3*D
#define M_TOT   16384         // B*S
#define T_TASKS 10
#define R_RANK  16
#define TAU_F   1.0f
#define EPS_F   1e-8f

// GEMM tiling
#define BM 128
#define BN 128
#define BK 16
#define LDP 20                // padded LDS row stride in floats (bank-conflict-free)
#define KTILES (D_MODEL / BK) // 48

typedef float v2f  __attribute__((ext_vector_type(2)));
typedef float v8f  __attribute__((ext_vector_type(8)));
typedef float f32x4 __attribute__((ext_vector_type(4)));
typedef int   v4i  __attribute__((vector_size(4 * sizeof(int))));

#if __has_builtin(__builtin_amdgcn_global_load_async_to_lds_b128) && \
    __has_builtin(__builtin_amdgcn_s_wait_asynccnt)
#define USE_ASYNC_LDS 1
#else
#define USE_ASYNC_LDS 0
#endif

#if USE_ASYNC_LDS
typedef __attribute__((address_space(1))) v4i gbl_v4i;
typedef __attribute__((address_space(3))) v4i lds_v4i;

// global -> LDS direct async copy, 16B per lane, tracked by ASYNCcnt
__device__ __forceinline__ void async_ld16(const float* g, float* l) {
  __builtin_amdgcn_global_load_async_to_lds_b128(
      (gbl_v4i*)(uintptr_t)g,
      (lds_v4i*)(uint32_t)(uintptr_t)l,
      0, 0);
}
#endif

// ---------------------------------------------------------------------------
// Kernel 1: sum-of-squares for each adapter tensor slice (40 reductions)
// ---------------------------------------------------------------------------
__global__ __launch_bounds__(256) void norms_kernel(
    const float* __restrict__ Aq, const float* __restrict__ Bq,
    const float* __restrict__ Av, const float* __restrict__ Bv,
    float* __restrict__ ss_out /*40*/) {
  const int b = blockIdx.x;
  const int tensor = b / T_TASKS;
  const int t = b % T_TASKS;
  const float* bases[4] = {Aq, Bq, Av, Bv};
  const int n = R_RANK * D_MODEL; // 12288
  const float* p = bases[tensor] + (size_t)t * n;

  float s = 0.0f;
  for (int i = threadIdx.x; i < n; i += 256) {
    float v = p[i];
    s += v * v;
  }
  __shared__ float red[256];
  red[threadIdx.x] = s;
  __syncthreads();
  for (int off = 128; off > 0; off >>= 1) {
    if (threadIdx.x < off) red[threadIdx.x] += red[threadIdx.x + off];
    __syncthreads();
  }
  if (threadIdx.x == 0) ss_out[b] = red[0];
}

// ---------------------------------------------------------------------------
// Kernel 2: coefficients  w_t = softmax(gate/tau)_t * alpha_t / (|A||B|+eps)
// ---------------------------------------------------------------------------
__global__ void coef_kernel(const float* __restrict__ gate,
                            const float* __restrict__ alpha,
                            const float* __restrict__ ss /*40*/,
                            float* __restrict__ wq /*10*/,
                            float* __restrict__ wv /*10*/) {
  const int t = threadIdx.x;
  if (t >= T_TASKS) return;
  float mx = -3.4e38f;
  for (int i = 0; i < T_TASKS; ++i) mx = fmaxf(mx, gate[i] / TAU_F);
  float sum = 0.0f;
  for (int i = 0; i < T_TASKS; ++i) sum += __expf(gate[i] / TAU_F - mx);
  const float beta = __expf(gate[t] / TAU_F - mx) / sum;
  const float coef = beta * alpha[t];
  wq[t] = coef / (sqrtf(ss[t])      * sqrtf(ss[10 + t]) + EPS_F);
  wv[t] = coef / (sqrtf(ss[20 + t]) * sqrtf(ss[30 + t]) + EPS_F);
}

// ---------------------------------------------------------------------------
// Kernel 3: W_eff[e,d] = qkv_w[e,d] + LoRA-q (e<D) + LoRA-v (e>=2D)
// ---------------------------------------------------------------------------
__global__ __launch_bounds__(256) void weff_kernel(
    const float* __restrict__ qkv_w,
    const float* __restrict__ Aq, const float* __restrict__ Bq,
    const float* __restrict__ Av, const float* __restrict__ Bv,
    const float* __restrict__ wq, const float* __restrict__ wv,
    float* __restrict__ W) {
  const int idx = blockIdx.x * 256 + threadIdx.x; // < 2304*768
  const int e = idx / D_MODEL;
  const int d = idx - e * D_MODEL;
  float val = qkv_w[idx];
  if (e < D_MODEL || e >= 2 * D_MODEL) {
    const bool isq = (e < D_MODEL);
    const float* A = isq ? Aq : Av;
    const float* Bm = isq ? Bq : Bv;
    const float* w = isq ? wq : wv;
    const int eo = isq ? e : (e - 2 * D_MODEL);
    float acc = 0.0f;
    #pragma unroll
    for (int t = 0; t < T_TASKS; ++t) {
      float s = 0.0f;
      const float* arow = A + (size_t)t * R_RANK * D_MODEL + d;     // stride D over r
      const float* brow = Bm + ((size_t)t * D_MODEL + eo) * R_RANK; // contiguous r
      #pragma unroll
      for (int r = 0; r < R_RANK; ++r) s = fmaf(arow[r * D_MODEL], brow[r], s);
      acc = fmaf(w[t], s, acc);
    }
    val += acc;
  }
  W[idx] = val;
}

// ---------------------------------------------------------------------------
// Kernel 4: out[m,n] = sum_k X[m,k]*W[n,k] + bias[n]
//   M=16384, N=2304, K=768.  fp32 WMMA 16x16x4, 128x128 WG tile, 8 waves,
//   each wave 64x32 (4x2 accumulators). Double-buffered LDS via async-to-LDS.
// ---------------------------------------------------------------------------
__global__ __launch_bounds__(256) void gemm_kernel(
    const float* __restrict__ X, const float* __restrict__ W,
    const float* __restrict__ bias, float* __restrict__ out) {
  __shared__ float xs[2][BM * LDP];
  __shared__ float wsh[2][BN * LDP];

  const int tid = threadIdx.x;
  const int wave = tid >> 5;
  const int lane = tid & 31;
  const int laneHalf = lane & 15;
  const int hi = lane >> 4;            // 0: K+0/1, 1: K+2/3 (A & B frag halves)
  const int wm = (wave >> 2) * 64;     // wave M offset in WG tile
  const int wn = (wave & 3) * 32;      // wave N offset in WG tile
  const int bm = blockIdx.y * BM;
  const int bn = blockIdx.x * BN;

  v8f acc[4][2];
  #pragma unroll
  for (int i = 0; i < 4; ++i)
    #pragma unroll
    for (int j = 0; j < 2; ++j) acc[i][j] = (v8f)0.0f;

#if USE_ASYNC_LDS
  // Each thread issues 2 async b128 copies per matrix per K-tile
  // (128x16 f32 tile = 512 x 16B). LDS dest 16B aligned: row*80 + {0,16,32,48}.
  #define GISSUE(K0, NBUF)                                                     \
    _Pragma("unroll")                                                          \
    for (int c = 0; c < 2; ++c) {                                              \
      const int l = c * 256 + tid;                                             \
      const int row = l >> 2;                                                  \
      const int c4 = (l & 3) << 2;                                             \
      async_ld16(X + (size_t)(bm + row) * D_MODEL + (K0) + c4,                 \
                 &xs[NBUF][row * LDP + c4]);                                   \
      async_ld16(W + (size_t)(bn + row) * D_MODEL + (K0) + c4,                 \
                 &wsh[NBUF][row * LDP + c4]);                                  \
    }

  GISSUE(0, 0);
  __builtin_amdgcn_s_wait_asynccnt(0);
  __syncthreads();
#else
  f32x4 xr[2], wr[2];
  #define GLOAD(K0)                                                            \
    _Pragma("unroll")                                                          \
    for (int c = 0; c < 2; ++c) {                                              \
      const int l = c * 256 + tid;                                             \
      const int row = l >> 2;                                                  \
      const int c4 = (l & 3) << 2;                                             \
      xr[c] = *(const f32x4*)(X + (size_t)(bm + row) * D_MODEL + (K0) + c4);   \
      wr[c] = *(const f32x4*)(W + (size_t)(bn + row) * D_MODEL + (K0) + c4);   \
    }
  #define LSTORE(BUF)                                                          \
    _Pragma("unroll")                                                          \
    for (int c = 0; c < 2; ++c) {                                              \
      const int l = c * 256 + tid;                                             \
      const int row = l >> 2;                                                  \
      const int c4 = (l & 3) << 2;                                             \
      v2f* px = (v2f*)&xs[BUF][row * LDP + c4];                                \
      px[0] = xr[c].lo; px[1] = xr[c].hi;                                      \
      v2f* pw = (v2f*)&wsh[BUF][row * LDP + c4];                               \
      pw[0] = wr[c].lo; pw[1] = wr[c].hi;                                      \
    }
  GLOAD(0);
  LSTORE(0);
  __syncthreads();
#endif

  for (int kt = 0; kt < KTILES; ++kt) {
    const int cur = kt & 1;
#if USE_ASYNC_LDS
    if (kt + 1 < KTILES) { GISSUE((kt + 1) * BK, cur ^ 1); }
#else
    if (kt + 1 < KTILES) { GLOAD((kt + 1) * BK); }
#endif

    // --- compute on buffer `cur` ---
    const float* xb = &xs[cur][(wm + laneHalf) * LDP + (hi << 1)];
    const float* wb = &wsh[cur][(wn + laneHalf) * LDP + (hi << 1)];
    #pragma unroll
    for (int kk = 0; kk < BK / 4; ++kk) {
      const int k = kk * 4;
      v2f a[4], b[2];
      #pragma unroll
      for (int i = 0; i < 4; ++i) a[i] = *(const v2f*)(xb + i * 16 * LDP + k);
      #pragma unroll
      for (int j = 0; j < 2; ++j) b[j] = *(const v2f*)(wb + j * 16 * LDP + k);
      #pragma unroll
      for (int i = 0; i < 4; ++i)
        #pragma unroll
        for (int j = 0; j < 2; ++j)
          acc[i][j] = __builtin_amdgcn_wmma_f32_16x16x4_f32(
              false, a[i], false, b[j], (short)0, acc[i][j], false, false);
    }

    if (kt + 1 < KTILES) {
#if USE_ASYNC_LDS
      __builtin_amdgcn_s_wait_asynccnt(0);  // my tile k+1 copies landed
      __syncthreads();                      // everyone's copies landed
#else
      __syncthreads();                      // everyone done reading buffer kt^1
      LSTORE((kt + 1) & 1);
      __syncthreads();
#endif
    }
  }
#if USE_ASYNC_LDS
  #undef GISSUE
#else
  #undef GLOAD
  #undef LSTORE
#endif

  // --- epilogue: bias + store (C layout: VGPR r -> rows M=r / M=r+8) ---
  const int n0 = bn + wn + laneHalf;
  #pragma unroll
  for (int j = 0; j < 2; ++j) {
    const int n = n0 + j * 16;
    const float bv = bias[n];
    #pragma unroll
    for (int i = 0; i < 4; ++i) {
      const int m = bm + wm + i * 16 + hi * 8;
      float* po = out + (size_t)m * N_OUT + n;
      #pragma unroll
      for (int r = 0; r < 8; ++r)
        po[(size_t)r * N_OUT] = acc[i][j][r] + bv;
    }
  }
}

// ---------------------------------------------------------------------------
extern "C" void kernel_launch(void* const* d_in, const int* in_sizes, int n_in,
                              void* d_out, int out_size, void* d_ws, size_t ws_size,
                              hipStream_t stream) {
  const float* x     = (const float*)d_in[0];
  const float* A_q   = (const float*)d_in[1];
  const float* B_q   = (const float*)d_in[2];
  const float* A_v   = (const float*)d_in[3];
  const float* B_v   = (const float*)d_in[4];
  const float* qkv_w = (const float*)d_in[5];
  const float* qkv_b = (const float*)d_in[6];
  const float* gate  = (const float*)d_in[7];
  const float* alpha = (const float*)d_in[8];
  float* out = (float*)d_out;

  float* wsf  = (float*)d_ws;
  float* ss   = wsf;        // [40]  sum-of-squares
  float* wq   = wsf + 40;   // [10]
  float* wv   = wsf + 50;   // [10]
  float* Weff = wsf + 64;   // [2304*768], 256B-aligned

  norms_kernel<<<40, 256, 0, stream>>>(A_q, B_q, A_v, B_v, ss);
  coef_kernel<<<1, 32, 0, stream>>>(gate, alpha, ss, wq, wv);
  weff_kernel<<<(N_OUT * D_MODEL) / 256, 256, 0, stream>>>(
      qkv_w, A_q, B_q, A_v, B_v, wq, wv, Weff);
  dim3 grid(N_OUT / BN, M_TOT / BM);
  gemm_kernel<<<grid, 256, 0, stream>>>(x, Weff, qkv_b, out);
}